// SpikeFP8Adder_Spatial_43860206027281
// MI455X (gfx1250) — compile-verified
//
#include <hip/hip_runtime.h>
#include <stdint.h>
#include <stddef.h>

// ---------------------------------------------------------------------------
// Gate-exact FP8(E4M3) "spatial adder": the reference's soft gates operate on
// exact {0,1} floats, so the network is exact Boolean logic. We pack the 8
// bits (MSB-first: sign, e3..e0, m2..m0) into one byte and evaluate the exact
// same datapath with integer ops (incl. the network's quirks: truncating
// round, exponent wrap mod 16, LZD saturating at 7, low nibble of the
// difference discarded before renormalization, full zero-squash on nz==0).
// ---------------------------------------------------------------------------

#define WAVE 32
#define BLOCK 256
#define WAVES_PER_BLOCK (BLOCK / WAVE)
#define BYTES_PER_ELEM 32                    // 8 f32 bits per operand
#define WAVE_SLICE (WAVE * BYTES_PER_ELEM)   // 1024 B per input per wave
#define LDS_PER_WAVE (2 * WAVE_SLICE)        // A slice + B slice

typedef float f32x4 __attribute__((ext_vector_type(4)));

// Builtin parameter types per the round-1 diagnostic:
//   param1: 'int __attribute__((vector_size(16))) __device__ *'  (AS1, non-const)
typedef int v4i __attribute__((vector_size(4 * sizeof(int))));
typedef __attribute__((address_space(1))) v4i* gv4i_t;   // global v4i*
typedef __attribute__((address_space(3))) v4i* lv4i_t;   // LDS v4i*

#if defined(__gfx1250__) && __has_builtin(__builtin_amdgcn_global_load_async_to_lds_b128)
#define USE_ASYNC_LDS 1
#else
#define USE_ASYNC_LDS 0
#endif

__device__ __forceinline__ unsigned fp8_gate_add(unsigned a, unsigned b) {
    // bit7 = sign, bits6..3 = exponent (e3..e0), bits2..0 = mantissa (m2..m0)
    const unsigned magA = a & 0x7Fu, magB = b & 0x7Fu;
    const bool age = magA >= magB;                 // gt|eq -> big = A on tie
    const unsigned big = age ? a : b;
    const unsigned sml = age ? b : a;
    const unsigned s_big = (big >> 7) & 1u;
    const unsigned e_big = (big >> 3) & 0xFu;
    const unsigned m_big = big & 7u;
    const unsigned s_sml = (sml >> 7) & 1u;
    const unsigned e_sml = (sml >> 3) & 0xFu;
    const unsigned m_sml = sml & 7u;
    const unsigned ediff = e_big - e_sml;          // exact, 0..15
    // 12-bit significands: [implicit(=|exp|!=0), m2..m0, 8 zero guard bits]
    const unsigned MB = ((e_big ? 1u : 0u) << 11) | (m_big << 8);
    const unsigned MS = ((((e_sml ? 1u : 0u) << 11) | (m_sml << 8)) >> ediff);
    const unsigned op_sub = s_big ^ s_sml;

    // ---- addition path (same signs) ----
    const unsigned sum   = MB + MS;
    const unsigned carry = (sum >> 12) & 1u;
    const unsigned top4_add = carry ? (8u | ((sum >> 9) & 7u)) : ((sum >> 8) & 0xFu);
    const unsigned exp_add  = (e_big + carry) & 0xFu;        // carry-out dropped

    // ---- subtraction path (opposite signs); MB >= MS guaranteed ----
    const unsigned diff = MB - MS;                 // 12-bit, non-negative
    const unsigned top8 = (diff >> 4) & 0xFFu;     // low nibble discarded (gate-exact)
    // LZD8 of the network: clz of top8 saturated at 7 (top8==0 -> 7).
    unsigned lzc = (unsigned)__builtin_clz((top8 << 1) | 1u) - 23u; // no clz(0)
    if (lzc > 7u) lzc = 7u;
    const unsigned left8    = (top8 << lzc) & 0xFFu;          // zeros shifted in
    const unsigned top4_sub = left8 >> 4;
    const unsigned exp_sub  = (e_big - lzc) & 0xFu;           // borrow dropped

    const unsigned top4  = op_sub ? top4_sub : top4_add;
    const unsigned exp_r = op_sub ? exp_sub  : exp_add;
    const unsigned nz    = (top4 >> 3) & 1u;                  // normalized MSB
    return nz ? ((s_big << 7) | (exp_r << 3) | (top4 & 7u)) : 0u;
}

__device__ __forceinline__ unsigned pack_bits(f32x4 v0, f32x4 v1) {
    return ((unsigned)(v0.x != 0.0f) << 7) | ((unsigned)(v0.y != 0.0f) << 6) |
           ((unsigned)(v0.z != 0.0f) << 5) | ((unsigned)(v0.w != 0.0f) << 4) |
           ((unsigned)(v1.x != 0.0f) << 3) | ((unsigned)(v1.y != 0.0f) << 2) |
           ((unsigned)(v1.z != 0.0f) << 1) |  (unsigned)(v1.w != 0.0f);
}

__global__ __launch_bounds__(BLOCK) void fp8_spatial_add_kernel(
    const float* __restrict__ A, const float* __restrict__ B,
    float* __restrict__ O, int nElems) {
#if USE_ASYNC_LDS
    __shared__ __align__(16) char smem[WAVES_PER_BLOCK * LDS_PER_WAVE];
#endif
    const int tid  = blockIdx.x * BLOCK + threadIdx.x;
    const int lane = threadIdx.x & (WAVE - 1);
    const bool valid = tid < nElems;

    f32x4 a0 = {0.f, 0.f, 0.f, 0.f}, a1 = a0, b0 = a0, b1 = a0;

#if USE_ASYNC_LDS
    const int wv = threadIdx.x >> 5;
    const int waveStart = tid - lane;
    char* lw = smem + wv * LDS_PER_WAVE;
    if (waveStart + WAVE <= nElems) {
        // CDNA5 async bulk path: each wave DMA-stages its private 1KB A/B
        // slices into LDS (512B per async b128), tracked by ASYNCcnt.
        gv4i_t ga = (gv4i_t)((const char*)A + (size_t)waveStart * BYTES_PER_ELEM + lane * 16);
        gv4i_t gb = (gv4i_t)((const char*)B + (size_t)waveStart * BYTES_PER_ELEM + lane * 16);
        lv4i_t la = (lv4i_t)(lw + lane * 16);
        lv4i_t lb = (lv4i_t)(lw + WAVE_SLICE + lane * 16);
        __builtin_amdgcn_global_load_async_to_lds_b128(ga, la, 0, 0);
        __builtin_amdgcn_global_load_async_to_lds_b128(ga, la, 512, 0);
        __builtin_amdgcn_global_load_async_to_lds_b128(gb, lb, 0, 0);
        __builtin_amdgcn_global_load_async_to_lds_b128(gb, lb, 512, 0);
#if __has_builtin(__builtin_amdgcn_s_wait_asynccnt)
        __builtin_amdgcn_s_wait_asynccnt(0);
#else
        asm volatile("s_wait_asynccnt 0" ::: "memory");
#endif
        // Slices are wave-private: no barrier needed.
        const f32x4* pa = (const f32x4*)(lw + lane * BYTES_PER_ELEM);
        const f32x4* pb = (const f32x4*)(lw + WAVE_SLICE + lane * BYTES_PER_ELEM);
        a0 = pa[0]; a1 = pa[1];
        b0 = pb[0]; b1 = pb[1];
    } else if (valid) {   // ragged tail: direct b128 loads
        const f32x4* pa = (const f32x4*)(A + (size_t)tid * 8);
        const f32x4* pb = (const f32x4*)(B + (size_t)tid * 8);
        a0 = pa[0]; a1 = pa[1];
        b0 = pb[0]; b1 = pb[1];
    }
#else
    if (valid) {
        const f32x4* pa = (const f32x4*)(A + (size_t)tid * 8);
        const f32x4* pb = (const f32x4*)(B + (size_t)tid * 8);
        a0 = pa[0]; a1 = pa[1];
        b0 = pb[0]; b1 = pb[1];
    }
#endif

    if (valid) {
        const unsigned r = fp8_gate_add(pack_bits(a0, a1), pack_bits(b0, b1));
        f32x4 o0, o1;
        o0.x = (float)((r >> 7) & 1u);
        o0.y = (float)((r >> 6) & 1u);
        o0.z = (float)((r >> 5) & 1u);
        o0.w = (float)((r >> 4) & 1u);
        o1.x = (float)((r >> 3) & 1u);
        o1.y = (float)((r >> 2) & 1u);
        o1.z = (float)((r >> 1) & 1u);
        o1.w = (float)( r        & 1u);
        f32x4* po = (f32x4*)(O + (size_t)tid * 8);
        // Output is write-once: non-temporal keeps it from evicting the
        // input stream out of the 192MB L2.
        __builtin_nontemporal_store(o0, po);
        __builtin_nontemporal_store(o1, po + 1);
    }
}

extern "C" void kernel_launch(void* const* d_in, const int* in_sizes, int n_in,
                              void* d_out, int out_size, void* d_ws, size_t ws_size,
                              hipStream_t stream) {
    (void)n_in; (void)out_size; (void)d_ws; (void)ws_size;
    const float* A = (const float*)d_in[0];
    const float* B = (const float*)d_in[1];
    float* O = (float*)d_out;
    const int nElems = in_sizes[0] / 8;          // (N, 8) bit-vectors
    const int blocks = (nElems + BLOCK - 1) / BLOCK;
    hipLaunchKernelGGL(fp8_spatial_add_kernel, dim3(blocks), dim3(BLOCK), 0, stream,
                       A, B, O, nElems);
}